// SupConLoss_HNM_16484084483424
// MI455X (gfx1250) — compile-verified
//
#include <hip/hip_runtime.h>
#include <math.h>

#define BB 2048
#define DD 512
#define INV_T (1.0f / 0.07f)

typedef __attribute__((ext_vector_type(2))) float    v2f;
typedef __attribute__((ext_vector_type(8))) float    v8f;
typedef __attribute__((ext_vector_type(4))) unsigned v4u;
typedef __attribute__((ext_vector_type(4))) int      v4i;
typedef __attribute__((ext_vector_type(8))) int      v8i;
typedef __attribute__((ext_vector_type(4))) float    f4;

#if defined(__has_builtin)
#if __has_builtin(__builtin_amdgcn_tensor_load_to_lds) && __has_builtin(__builtin_amdgcn_s_wait_tensorcnt)
#define HAVE_TDM 1
#endif
#endif

struct Ctrl {
    unsigned n_neg;
    unsigned remaining;
    unsigned prefix;
    float    threshold;
};

// ---- float <-> order-preserving key ----
__device__ __forceinline__ unsigned f2k(float f) {
    unsigned u = __float_as_uint(f);
    return (u & 0x80000000u) ? ~u : (u | 0x80000000u);
}
__device__ __forceinline__ float k2f(unsigned k) {
    unsigned u = (k & 0x80000000u) ? (k & 0x7FFFFFFFu) : ~k;
    return __uint_as_float(u);
}

// ============================================================
// Kernel 0: zero scratch (hist + ctrl)
// ============================================================
__global__ __launch_bounds__(256) void init_scratch(unsigned* hist, Ctrl* ctrl) {
    hist[threadIdx.x] = 0u;
    if (threadIdx.x == 0) { ctrl->n_neg = 0u; ctrl->remaining = 0u; ctrl->prefix = 0u; ctrl->threshold = 0.f; }
}

// ============================================================
// GEMM tiling constants
// ============================================================
#define BLK_ROWS 64
#define BLK_COLS 128
#define KC 32
#define NCHUNK (DD / KC)     // 16
#define ROWPAD 34            // KC + 2 dwords: TDM pad makes ds reads bank-conflict-free

#if HAVE_TDM
// Issue one TDM 2-D tile load: `rows` rows x KC floats from F (row stride DD)
// into LDS at byte offset ldsOff, padding each 32-dword row by 2 dwords.
__device__ __forceinline__ void tdm_load_panel(unsigned ldsOff, const float* gptr, int rows) {
    unsigned long long ga = (unsigned long long)(uintptr_t)gptr;
    v4u g0;
    g0[0] = 1u;                                            // count=1, user mode, no gather
    g0[1] = ldsOff;                                        // lds_addr (bytes)
    g0[2] = (unsigned)(ga & 0xFFFFFFFFull);                // global_addr[31:0]
    g0[3] = (unsigned)((ga >> 32) & 0x01FFFFFFull)
          | (2u << 30);                                    // global_addr[56:32] | type=2
    v8i g1;
    g1[0] = (int)((2u << 16)      // data_size = 4 bytes
                | (1u << 20)      // pad_enable
                | (4u << 22)      // pad_interval: 2^(4+1)=32 dwords
                | (1u << 25));    // pad_amount: 2 dwords
    g1[1] = (int)((unsigned)DD << 16);                     // tensor_dim0 = 512 (low16 @ [31:16])
    g1[2] = (int)((unsigned)BB << 16);                     // tensor_dim0 hi=0 | tensor_dim1 lo16
    g1[3] = (int)((unsigned)KC << 16);                     // tensor_dim1 hi=0 | tile_dim0 = 32
    g1[4] = rows;                                          // tile_dim1 = rows, tile_dim2 = 0
    g1[5] = DD;                                            // tensor_dim0_stride = 512
    g1[6] = 0;
    g1[7] = 0;
    v4i z4 = {0, 0, 0, 0};
#if __clang_major__ >= 23
    v8i z8 = {0, 0, 0, 0, 0, 0, 0, 0};
    __builtin_amdgcn_tensor_load_to_lds(g0, g1, z4, z4, z8, 0);
#else
    __builtin_amdgcn_tensor_load_to_lds(g0, g1, z4, z4, 0);
#endif
}
#endif

// ============================================================
// Kernel 1: S = (F F^T) / T via V_WMMA_F32_16X16X4_F32.
// Block 256 thr (8 waves), tile 64x128. K staged in LDS in KC=32 chunks:
//  - TDM path: double-buffered tensor_load_to_lds issued by wave 0,
//    synced with s_wait_tensorcnt + workgroup barrier.
//  - Fallback: cooperative float4 global->LDS staging.
// Wave tile: 16 rows x 64 cols (4 accumulators, A fragment reused 4x).
// ============================================================
__global__ __launch_bounds__(256) void gram_wmma(const float* __restrict__ F,
                                                 float* __restrict__ S) {
    __shared__ float ldsA[2][BLK_ROWS * ROWPAD];
    __shared__ float ldsB[2][BLK_COLS * ROWPAD];

    const int tid  = threadIdx.x;
    const int lane = tid & 31;
    const int wave = tid >> 5;
    const int m    = lane & 15;   // M (A rows) / N (B cols) lane index
    const int h    = lane >> 4;   // lane-half selects K sub-pair

    const int rowL = (wave >> 1) * 16;   // 0..48 within block tile
    const int colL = (wave & 1) * 64;    // 0 or 64 within block tile
    const int row0 = blockIdx.y * BLK_ROWS + rowL;
    const int col0 = blockIdx.x * BLK_COLS + colL;

    const float* gA = F + (size_t)blockIdx.y * BLK_ROWS * DD;   // A panel rows
    const float* gB = F + (size_t)blockIdx.x * BLK_COLS * DD;   // B panel rows

    v8f acc0 = {}; v8f acc1 = {}; v8f acc2 = {}; v8f acc3 = {};

#if HAVE_TDM
    const unsigned ldsAoff[2] = { (unsigned)(uintptr_t)&ldsA[0][0], (unsigned)(uintptr_t)&ldsA[1][0] };
    const unsigned ldsBoff[2] = { (unsigned)(uintptr_t)&ldsB[0][0], (unsigned)(uintptr_t)&ldsB[1][0] };
    if (wave == 0) {
        tdm_load_panel(ldsAoff[0], gA, BLK_ROWS);
        tdm_load_panel(ldsBoff[0], gB, BLK_COLS);
    }
#endif

    for (int c = 0; c < NCHUNK; ++c) {
        const int buf = c & 1;
#if HAVE_TDM
        if (wave == 0) __builtin_amdgcn_s_wait_tensorcnt(0);   // chunk c resident in LDS
        __syncthreads();                                       // publish LDS + retire old-buf readers
        if (wave == 0 && c + 1 < NCHUNK) {                     // prefetch chunk c+1 into other buffer
            tdm_load_panel(ldsAoff[buf ^ 1], gA + (size_t)(c + 1) * KC, BLK_ROWS);
            tdm_load_panel(ldsBoff[buf ^ 1], gB + (size_t)(c + 1) * KC, BLK_COLS);
        }
#else
        // Cooperative staging: A 64x32 (512 float4), B 128x32 (1024 float4)
        {
            const float* srcA = gA + (size_t)c * KC;
            const float* srcB = gB + (size_t)c * KC;
            __syncthreads();  // retire previous chunk's readers before overwrite
            for (int v = tid; v < (BLK_ROWS * KC) / 4; v += 256) {
                const int r = v >> 3, kv = (v & 7) * 4;
                f4 d = *(const f4*)(srcA + (size_t)r * DD + kv);
                *(f4*)&ldsA[buf][r * ROWPAD + kv] = d;
            }
            for (int v = tid; v < (BLK_COLS * KC) / 4; v += 256) {
                const int r = v >> 3, kv = (v & 7) * 4;
                f4 d = *(const f4*)(srcB + (size_t)r * DD + kv);
                *(f4*)&ldsB[buf][r * ROWPAD + kv] = d;
            }
            __syncthreads();
        }
#endif
        const float* As = &ldsA[buf][0];
        const float* Bs = &ldsB[buf][0];
#pragma unroll
        for (int k = 0; k < KC; k += 4) {
            const int off = k + 2 * h;
            v2f a   = *(const v2f*)(As + (rowL + m) * ROWPAD + off);
            v2f vb0 = *(const v2f*)(Bs + (colL +  0 + m) * ROWPAD + off);
            v2f vb1 = *(const v2f*)(Bs + (colL + 16 + m) * ROWPAD + off);
            v2f vb2 = *(const v2f*)(Bs + (colL + 32 + m) * ROWPAD + off);
            v2f vb3 = *(const v2f*)(Bs + (colL + 48 + m) * ROWPAD + off);
            acc0 = __builtin_amdgcn_wmma_f32_16x16x4_f32(false, a, false, vb0, (short)0, acc0, false, false);
            acc1 = __builtin_amdgcn_wmma_f32_16x16x4_f32(false, a, false, vb1, (short)0, acc1, false, false);
            acc2 = __builtin_amdgcn_wmma_f32_16x16x4_f32(false, a, false, vb2, (short)0, acc2, false, false);
            acc3 = __builtin_amdgcn_wmma_f32_16x16x4_f32(false, a, false, vb3, (short)0, acc3, false, false);
        }
    }

    // C/D layout: lane l, VGPR v -> row = v + 8*(l>>4), col = l&15
#pragma unroll
    for (int v = 0; v < 8; ++v) {
        const int row = row0 + v + 8 * h;
        float* out = S + (size_t)row * BB + col0 + m;
        out[0]  = acc0[v] * INV_T;
        out[16] = acc1[v] * INV_T;
        out[32] = acc2[v] * INV_T;
        out[48] = acc3[v] * INV_T;
    }
}

// ============================================================
// Kernel 2: row max
// ============================================================
__global__ __launch_bounds__(256) void row_max_k(const float* __restrict__ S,
                                                 float* __restrict__ rowmax) {
    const int i = blockIdx.x;
    float mx = -INFINITY;
    for (int j = threadIdx.x; j < BB; j += 256) mx = fmaxf(mx, S[(size_t)i * BB + j]);
    __shared__ float sm[256];
    sm[threadIdx.x] = mx;
    __syncthreads();
    for (int s = 128; s > 0; s >>= 1) {
        if (threadIdx.x < s) sm[threadIdx.x] = fmaxf(sm[threadIdx.x], sm[threadIdx.x + s]);
        __syncthreads();
    }
    if (threadIdx.x == 0) rowmax[i] = sm[0];
}

// ============================================================
// Kernel 3: radix-select histogram pass (8 bits / pass).
// ============================================================
__global__ __launch_bounds__(256) void hist_pass(const float* __restrict__ S,
                                                 const float* __restrict__ rowmax,
                                                 const int* __restrict__ labels,
                                                 unsigned* __restrict__ hist,
                                                 Ctrl* __restrict__ ctrl,
                                                 int shift, unsigned maskHi, int firstPass) {
    __shared__ unsigned lh[256];
    __shared__ unsigned lneg;
    lh[threadIdx.x] = 0u;
    if (threadIdx.x == 0) lneg = 0u;
    __syncthreads();

    const unsigned prefix = firstPass ? 0u : ctrl->prefix;
    const int i0 = blockIdx.x * 2;
    unsigned localNeg = 0;

    for (int r = 0; r < 2; ++r) {
        const int i  = i0 + r;
        const int li = labels[i];
        const float rm = rowmax[i];
        for (int j = threadIdx.x; j < BB; j += 256) {
            if (j == i) continue;
            if (labels[j] == li) continue;
            ++localNeg;
            const float logit = S[(size_t)i * BB + j] - rm;
            const unsigned key = f2k(logit);
            if ((key & maskHi) == prefix) atomicAdd(&lh[(key >> shift) & 0xFFu], 1u);
        }
    }
    if (firstPass && localNeg) atomicAdd(&lneg, localNeg);
    __syncthreads();
    if (lh[threadIdx.x]) atomicAdd(&hist[threadIdx.x], lh[threadIdx.x]);
    if (firstPass && threadIdx.x == 0 && lneg) atomicAdd(&ctrl->n_neg, lneg);
}

// ============================================================
// Kernel 4: radix-select digit pick (single block), zero hist for next pass
// ============================================================
__global__ __launch_bounds__(256) void select_pass(unsigned* __restrict__ hist,
                                                   Ctrl* __restrict__ ctrl,
                                                   int shift, int firstPass, int lastPass) {
    __shared__ unsigned h[256];
    h[threadIdx.x] = hist[threadIdx.x];
    __syncthreads();
    if (threadIdx.x == 0) {
        unsigned remaining;
        if (firstPass) {
            unsigned k = (unsigned)(0.5f * (float)ctrl->n_neg);   // n_neg < 2^23: exact
            if (k < 1u) k = 1u;
            remaining = k;
        } else {
            remaining = ctrl->remaining;
        }
        int d = 255;
        for (; d >= 0; --d) {
            const unsigned c = h[d];
            if (c >= remaining) break;
            remaining -= c;
        }
        if (d < 0) d = 0;
        ctrl->remaining = remaining;
        const unsigned pfx = (firstPass ? 0u : ctrl->prefix) | ((unsigned)d << shift);
        ctrl->prefix = pfx;
        if (lastPass) ctrl->threshold = k2f(pfx);   // exact k-th largest negative logit
    }
    __syncthreads();
    hist[threadIdx.x] = 0u;
}

// ============================================================
// Kernel 5: masks + per-row softmax stats; S is read and overwritten
// in-place by the hard-negative mask (same thread, read-before-write).
// (final_mask*log_prob).sum == ml - rp*log(se+1e-12)
// ============================================================
__global__ __launch_bounds__(256) void finalize_rows(float* __restrict__ hardS,
                                                     float* __restrict__ finalMask,
                                                     const float* __restrict__ rowmax,
                                                     const int* __restrict__ labels,
                                                     const Ctrl* __restrict__ ctrl,
                                                     float* __restrict__ lossVec,
                                                     float* __restrict__ singleVec) {
    const int i  = blockIdx.x;
    const int li = labels[i];
    const float rm  = rowmax[i];
    const float thr = ctrl->threshold;

    float se = 0.f, rp = 0.f, ml = 0.f;
    for (int j = threadIdx.x; j < BB; j += 256) {
        const size_t idx = (size_t)i * BB + j;
        const float logit = hardS[idx] - rm;
        const bool diag = (j == i);
        const bool pos  = (labels[j] == li);
        const bool neg  = (!pos) && (!diag);
        const bool hard = neg && (logit >= thr);
        const bool fm   = (pos || hard) && !diag;
        hardS[idx]     = hard ? 1.f : 0.f;
        finalMask[idx] = fm ? 1.f : 0.f;
        if (!diag) se += __expf(logit);
        if (fm) { rp += 1.f; ml += logit; }
    }

    __shared__ float s0[256], s1[256], s2[256];
    s0[threadIdx.x] = se; s1[threadIdx.x] = rp; s2[threadIdx.x] = ml;
    __syncthreads();
    for (int s = 128; s > 0; s >>= 1) {
        if (threadIdx.x < s) {
            s0[threadIdx.x] += s0[threadIdx.x + s];
            s1[threadIdx.x] += s1[threadIdx.x + s];
            s2[threadIdx.x] += s2[threadIdx.x + s];
        }
        __syncthreads();
    }
    if (threadIdx.x == 0) {
        const float seT = s0[0], rpT = s1[0], mlT = s2[0];
        const float single = (rpT == 0.f) ? 1.f : 0.f;
        const float mlpp = (mlT - rpT * logf(seT + 1e-12f)) / (rpT + single);
        lossVec[i]   = -mlpp * (1.f - single);
        singleVec[i] = single;
    }
}

// ============================================================
// Kernel 6: final scalar loss
// ============================================================
__global__ __launch_bounds__(256) void final_loss(const float* __restrict__ lossVec,
                                                  const float* __restrict__ singleVec,
                                                  float* __restrict__ out) {
    float ls = 0.f, ss = 0.f;
    for (int i = threadIdx.x; i < BB; i += 256) { ls += lossVec[i]; ss += singleVec[i]; }
    __shared__ float a[256], b[256];
    a[threadIdx.x] = ls; b[threadIdx.x] = ss;
    __syncthreads();
    for (int s = 128; s > 0; s >>= 1) {
        if (threadIdx.x < s) { a[threadIdx.x] += a[threadIdx.x + s]; b[threadIdx.x] += b[threadIdx.x + s]; }
        __syncthreads();
    }
    if (threadIdx.x == 0) out[0] = a[0] / ((float)BB - b[0]);
}

// ============================================================
extern "C" void kernel_launch(void* const* d_in, const int* in_sizes, int n_in,
                              void* d_out, int out_size, void* d_ws, size_t ws_size,
                              hipStream_t stream) {
    const float* F      = (const float*)d_in[0];
    const int*   labels = (const int*)d_in[1];

    float* out   = (float*)d_out;
    float* hardS = out + 1;                       // holds S, then hard_negatives_mask
    float* fmask = out + 1 + (size_t)BB * BB;     // final_mask

    float*    rowmax    = (float*)d_ws;
    float*    lossVec   = rowmax + BB;
    float*    singleVec = lossVec + BB;
    unsigned* hist      = (unsigned*)(singleVec + BB);
    Ctrl*     ctrl      = (Ctrl*)(hist + 256);

    init_scratch<<<1, 256, 0, stream>>>(hist, ctrl);

    gram_wmma<<<dim3(BB / BLK_COLS, BB / BLK_ROWS), 256, 0, stream>>>(F, hardS);

    row_max_k<<<BB, 256, 0, stream>>>(hardS, rowmax);

    const int shifts[4] = {24, 16, 8, 0};
    for (int p = 0; p < 4; ++p) {
        const int sh = shifts[p];
        const unsigned maskHi = (p == 0) ? 0u : (0xFFFFFFFFu << (sh + 8));
        hist_pass<<<BB / 2, 256, 0, stream>>>(hardS, rowmax, labels, hist, ctrl,
                                              sh, maskHi, p == 0 ? 1 : 0);
        select_pass<<<1, 256, 0, stream>>>(hist, ctrl, sh, p == 0 ? 1 : 0, p == 3 ? 1 : 0);
    }

    finalize_rows<<<BB, 256, 0, stream>>>(hardS, fmask, rowmax, labels, ctrl,
                                          lossVec, singleVec);

    final_loss<<<1, 256, 0, stream>>>(lossVec, singleVec, out);
}